// AttnDecoderRNN_72868415144086
// MI455X (gfx1250) — compile-verified
//
#include <hip/hip_runtime.h>
#include <math.h>

#define H 1024
#define V 50257
#define L 512

typedef float v2f __attribute__((ext_vector_type(2)));
typedef float v8f __attribute__((ext_vector_type(8)));

__device__ __forceinline__ float wred(float v) {
#pragma unroll
  for (int o = 16; o > 0; o >>= 1) v += __shfl_down(v, o, 32);
  return v;
}

// ---- stage 1: scores[l] = dot(concat(e, h0), attn_w[l]) + attn_b[l] --------
__global__ void k_scores(const int* __restrict__ inp, const float* __restrict__ emb,
                         const float* __restrict__ hid, const float* __restrict__ attn_w,
                         const float* __restrict__ attn_b, float* __restrict__ scores) {
  const int lane = threadIdx.x & 31;
  const int l = (blockIdx.x * blockDim.x + threadIdx.x) >> 5;
  if (l >= L) return;
  const float* e = emb + (size_t)inp[0] * H;
  const float* w = attn_w + (size_t)l * (2 * H);
  float acc = 0.f;
#pragma unroll 4
  for (int j = lane * 4; j < 2 * H; j += 128) {
    float4 wv = *(const float4*)(w + j);
    float4 vv = (j < H) ? *(const float4*)(e + j) : *(const float4*)(hid + (j - H));
    acc += wv.x * vv.x + wv.y * vv.y + wv.z * vv.z + wv.w * vv.w;
  }
  acc = wred(acc);
  if (lane == 0) scores[l] = acc + attn_b[l];
}

// ---- stage 2: softmax over L=512 scores ------------------------------------
__global__ void k_softmax_attn(const float* __restrict__ scores,
                               float* __restrict__ aw_ws, float* __restrict__ aw_out) {
  __shared__ float red[L];
  const int t = threadIdx.x;
  float v = scores[t];
  red[t] = v;
  __syncthreads();
  for (int s = L / 2; s > 0; s >>= 1) {
    if (t < s) red[t] = fmaxf(red[t], red[t + s]);
    __syncthreads();
  }
  float m = red[0];
  __syncthreads();
  float e = expf(v - m);
  red[t] = e;
  __syncthreads();
  for (int s = L / 2; s > 0; s >>= 1) {
    if (t < s) red[t] += red[t + s];
    __syncthreads();
  }
  float w = e / red[0];
  aw_ws[t] = w;
  aw_out[t] = w;
}

// ---- stage 3: attn_applied[h] = sum_l aw[l] * enc[l][h] --------------------
__global__ void k_attn_apply(const float* __restrict__ aw, const float* __restrict__ enc,
                             float* __restrict__ attn) {
  __shared__ float w[L];
  const int t = threadIdx.x;
  for (int l = t; l < L; l += blockDim.x) w[l] = aw[l];
  __syncthreads();
  const int hcol = blockIdx.x * blockDim.x + t;
  float acc = 0.f;
#pragma unroll 8
  for (int l = 0; l < L; ++l) acc += w[l] * enc[(size_t)l * H + hcol];
  attn[hcol] = acc;
}

// ---- stage 4: x = relu(comb_w @ concat(e, attn_applied) + comb_b) ----------
__global__ void k_combine(const int* __restrict__ inp, const float* __restrict__ emb,
                          const float* __restrict__ attn, const float* __restrict__ comb_w,
                          const float* __restrict__ comb_b, float* __restrict__ x) {
  const int lane = threadIdx.x & 31;
  const int i = (blockIdx.x * blockDim.x + threadIdx.x) >> 5;
  if (i >= H) return;
  const float* e = emb + (size_t)inp[0] * H;
  const float* w = comb_w + (size_t)i * (2 * H);
  float acc = 0.f;
#pragma unroll 4
  for (int j = lane * 4; j < 2 * H; j += 128) {
    float4 wv = *(const float4*)(w + j);
    float4 vv = (j < H) ? *(const float4*)(e + j) : *(const float4*)(attn + (j - H));
    acc += wv.x * vv.x + wv.y * vv.y + wv.z * vv.z + wv.w * vv.w;
  }
  acc = wred(acc);
  if (lane == 0) x[i] = fmaxf(acc + comb_b[i], 0.f);
}

// ---- stage 5: fused GRU step (r,z,n gate layout) ---------------------------
__global__ void k_gru(const float* __restrict__ x, const float* __restrict__ h0,
                      const float* __restrict__ w_ih, const float* __restrict__ w_hh,
                      const float* __restrict__ b_ih, const float* __restrict__ b_hh,
                      float* __restrict__ h_ws, float* __restrict__ h_out) {
  const int lane = threadIdx.x & 31;
  const int i = (blockIdx.x * blockDim.x + threadIdx.x) >> 5;
  if (i >= H) return;
  const float* wr = w_ih + (size_t)i * H;
  const float* wz = w_ih + (size_t)(H + i) * H;
  const float* wn = w_ih + (size_t)(2 * H + i) * H;
  const float* ur = w_hh + (size_t)i * H;
  const float* uz = w_hh + (size_t)(H + i) * H;
  const float* un = w_hh + (size_t)(2 * H + i) * H;
  float ar = 0.f, az = 0.f, an = 0.f, br = 0.f, bz = 0.f, bn = 0.f;
#pragma unroll 2
  for (int j = lane * 4; j < H; j += 128) {
    float4 xv = *(const float4*)(x + j);
    float4 hv = *(const float4*)(h0 + j);
    float4 w;
    w = *(const float4*)(wr + j); ar += w.x * xv.x + w.y * xv.y + w.z * xv.z + w.w * xv.w;
    w = *(const float4*)(wz + j); az += w.x * xv.x + w.y * xv.y + w.z * xv.z + w.w * xv.w;
    w = *(const float4*)(wn + j); an += w.x * xv.x + w.y * xv.y + w.z * xv.z + w.w * xv.w;
    w = *(const float4*)(ur + j); br += w.x * hv.x + w.y * hv.y + w.z * hv.z + w.w * hv.w;
    w = *(const float4*)(uz + j); bz += w.x * hv.x + w.y * hv.y + w.z * hv.z + w.w * hv.w;
    w = *(const float4*)(un + j); bn += w.x * hv.x + w.y * hv.y + w.z * hv.z + w.w * hv.w;
  }
  ar = wred(ar); az = wred(az); an = wred(an);
  br = wred(br); bz = wred(bz); bn = wred(bn);
  if (lane == 0) {
    float r = 1.f / (1.f + expf(-(ar + b_ih[i] + br + b_hh[i])));
    float z = 1.f / (1.f + expf(-(az + b_ih[H + i] + bz + b_hh[H + i])));
    float n = tanhf(an + b_ih[2 * H + i] + r * (bn + b_hh[2 * H + i]));
    float hn = (1.f - z) * n + z * h0[i];
    h_ws[i] = hn;
    h_out[i] = hn;
  }
}

// ---- stage 6: logits = out_w @ h_new + out_b, via V_WMMA_F32_16X16X4_F32 ---
// One wave per 16-row tile. A = 16x4 fp32 slab of out_w (ISA layout: lane l<16
// holds row l, K0/K1; lane l+16 holds row l, K2/K3). B[k][n] = h[k] broadcast
// over all N columns, so every D column holds the GEMV result; C chains the
// fp32 accumulator over the K loop (256 WMMAs / tile).
__global__ void k_logits_wmma(const float* __restrict__ out_w,
                              const float* __restrict__ out_b,
                              const float* __restrict__ h,
                              float* __restrict__ logits) {
  const int lane = threadIdx.x & 31;
  const int tile = blockIdx.x * (blockDim.x >> 5) + (threadIdx.x >> 5);
  const int rowBase = tile * 16;
  if (rowBase >= V) return;  // wave-uniform exit; EXEC stays all-1s for WMMA

  int r = rowBase + (lane & 15);
  r = (r < V) ? r : (V - 1);              // clamp tail rows for loads
  const int koff = (lane < 16) ? 0 : 2;   // K half selected by lane group
  const float* arow = out_w + (size_t)r * H;

  v8f c = {0.f, 0.f, 0.f, 0.f, 0.f, 0.f, 0.f, 0.f};
  for (int kb = 0; kb < H; kb += 4) {
    v2f a = *(const v2f*)(arow + kb + koff);
    v2f b = *(const v2f*)(h + kb + koff);  // broadcast: same value for all N
    c = __builtin_amdgcn_wmma_f32_16x16x4_f32(
        /*neg_a=*/false, a, /*neg_b=*/false, b,
        /*c_mod=*/(short)0, c, /*reuse_a=*/false, /*reuse_b=*/false);
  }
  // D layout: VGPR r0 <-> row r0 + 8*(lane>=16), all columns identical.
  if ((lane & 15) == 0) {
    const int base = rowBase + ((lane >> 4) << 3);
#pragma unroll
    for (int q = 0; q < 8; ++q) {
      const int row = base + q;
      if (row < V) logits[row] = c[q] + out_b[row];
    }
  }
}

// ---- stage 7: log_softmax over V logits ------------------------------------
__global__ void k_logsoftmax(const float* __restrict__ logits, float* __restrict__ out) {
  __shared__ float red[1024];
  const int t = threadIdx.x;
  float m = -INFINITY;
  for (int v = t; v < V; v += 1024) m = fmaxf(m, logits[v]);
  red[t] = m;
  __syncthreads();
  for (int s = 512; s > 0; s >>= 1) {
    if (t < s) red[t] = fmaxf(red[t], red[t + s]);
    __syncthreads();
  }
  m = red[0];
  __syncthreads();
  float sum = 0.f;
  for (int v = t; v < V; v += 1024) sum += expf(logits[v] - m);
  red[t] = sum;
  __syncthreads();
  for (int s = 512; s > 0; s >>= 1) {
    if (t < s) red[t] += red[t + s];
    __syncthreads();
  }
  const float lz = m + logf(red[0]);
  for (int v = t; v < V; v += 1024) out[v] = logits[v] - lz;
}

extern "C" void kernel_launch(void* const* d_in, const int* in_sizes, int n_in,
                              void* d_out, int out_size, void* d_ws, size_t ws_size,
                              hipStream_t stream) {
  const int*   inp    = (const int*)  d_in[0];
  const float* hidden = (const float*)d_in[1];
  const float* enc    = (const float*)d_in[2];
  const float* emb    = (const float*)d_in[3];
  const float* attn_w = (const float*)d_in[4];
  const float* attn_b = (const float*)d_in[5];
  const float* comb_w = (const float*)d_in[6];
  const float* comb_b = (const float*)d_in[7];
  const float* w_ih   = (const float*)d_in[8];
  const float* w_hh   = (const float*)d_in[9];
  const float* b_ih   = (const float*)d_in[10];
  const float* b_hh   = (const float*)d_in[11];
  const float* out_w  = (const float*)d_in[12];
  const float* out_b  = (const float*)d_in[13];

  // d_out layout: [0,V) log_softmax | [V,V+H) h_new | [V+H,V+H+L) attn_weights
  float* out = (float*)d_out;
  float* ws  = (float*)d_ws;
  float* scores = ws;            // 512
  float* aw     = ws + 512;      // 512
  float* attn   = ws + 1024;     // 1024
  float* xbuf   = ws + 2048;     // 1024
  float* hnew   = ws + 3072;     // 1024
  float* logits = ws + 4096;     // 50257

  k_scores      <<<L / 8,           256,  0, stream>>>(inp, emb, hidden, attn_w, attn_b, scores);
  k_softmax_attn<<<1,               L,    0, stream>>>(scores, aw, out + V + H);
  k_attn_apply  <<<H / 256,         256,  0, stream>>>(aw, enc, attn);
  k_combine     <<<H / 8,           256,  0, stream>>>(inp, emb, attn, comb_w, comb_b, xbuf);
  k_gru         <<<H / 8,           256,  0, stream>>>(xbuf, hidden, w_ih, w_hh, b_ih, b_hh, hnew, out + V);
  k_logits_wmma <<<(V + 127) / 128, 256,  0, stream>>>(out_w, out_b, hnew, logits);
  k_logsoftmax  <<<1,               1024, 0, stream>>>(logits, out);
}